// HybridRegression_65481071407194
// MI455X (gfx1250) — compile-verified
//
#include <hip/hip_runtime.h>

// out[r] = sum_k cos(x[r,k]) * w[k] + b   for B = out_size rows, W = 32.
// Memory-bound (132 MB @ 23.3 TB/s ~ 5.7us); dot product done with exact
// fp32 WMMA (v_wmma_f32_16x16x4_f32) accumulating C in f32.

typedef __attribute__((ext_vector_type(2))) float v2f;
typedef __attribute__((ext_vector_type(8))) float v8f;

__global__ __launch_bounds__(256) void hybrid_cos_dot_kernel(
    const float* __restrict__ x,     // [B, 32] row-major, rows are 128B-aligned cachelines
    const float* __restrict__ w,     // [32]
    const float* __restrict__ bias,  // [1]
    float* __restrict__ out,         // [B]
    int ntiles)                      // B / 16
{
    const int lane  = threadIdx.x & 31;
    const int waveInBlock = threadIdx.x >> 5;
    const int tile  = blockIdx.x * 8 + waveInBlock;   // one wave32 per 16-row tile
    if (tile >= ntiles) return;                       // wave-uniform: EXEC stays all-ones

    const int row   = lane & 15;   // A-matrix M for this lane
    const int khalf = lane >> 4;   // 0: K={0,1}+4c ; 1: K={2,3}+4c  (ISA 16x4 f32 A layout)
    const int r0    = tile << 4;

    // C initialized with the bias: D = A*B + C gives sum + b directly.
    const float bval = bias[0];
    v8f acc;
#pragma unroll
    for (int i = 0; i < 8; ++i) acc[i] = bval;

    const float* xrow = x + (size_t)(r0 + row) * 32 + 2 * khalf;  // 8B aligned
    const float* wp   = w + 2 * khalf;

#pragma unroll
    for (int c = 0; c < 8; ++c) {            // K = 32 in 8 chunks of 4
        v2f xa = *(const v2f*)(xrow + 4 * c);   // global_load_b64
        v2f wb = *(const v2f*)(wp   + 4 * c);   // B[k,n] = w[k] replicated over n
        v2f a;
        a[0] = __cosf(xa[0]);                   // v_cos_f32 path
        a[1] = __cosf(xa[1]);
        // (neg_a, A, neg_b, B, c_mod, C, reuse_a, reuse_b)
        acc = __builtin_amdgcn_wmma_f32_16x16x4_f32(
            false, a, false, wb, (short)0, acc, false, false);
    }

    // Every column of D is identical. Lanes 0..15: VGPR v = D[v, lane] (rows 0..7);
    // lanes 16..31: VGPR v = D[v+8, lane-16] (rows 8..15). Lane 0 and lane 16 store.
    if (row == 0) {
        float4 lo = make_float4(acc[0], acc[1], acc[2], acc[3]);
        float4 hi = make_float4(acc[4], acc[5], acc[6], acc[7]);
        float* o = out + r0 + khalf * 8;        // 32B aligned
        *(float4*)(o)     = lo;                 // global_store_b128
        *(float4*)(o + 4) = hi;                 // global_store_b128
    }
}

extern "C" void kernel_launch(void* const* d_in, const int* in_sizes, int n_in,
                              void* d_out, int out_size, void* d_ws, size_t ws_size,
                              hipStream_t stream) {
    // setup_inputs order: x [B,32], theta [32] (unused: RZ leaves <Z> invariant),
    // w [1,32], b [1]
    const float* x = (const float*)d_in[0];
    const float* w = (const float*)d_in[2];
    const float* b = (const float*)d_in[3];
    float* out = (float*)d_out;

    const int ntiles = out_size >> 4;            // BATCH is a multiple of 16
    const int blocks = (ntiles + 7) / 8;         // 8 wave32 per 256-thread block
    hipLaunchKernelGGL(hybrid_cos_dot_kernel, dim3(blocks), dim3(256), 0, stream,
                       x, w, b, out, ntiles);
}